// inv_comp_41695542509653
// MI455X (gfx1250) — compile-verified
//
#include <hip/hip_runtime.h>

typedef float v2f __attribute__((ext_vector_type(2)));
typedef float v8f __attribute__((ext_vector_type(8)));

// Fast softplus on raw CDNA5 trans ops.
// softplus(x) = max(x,0) + ln(1 + exp(-|x|)); the log argument is in (1,2],
// the exp argument is <= 0, so no denorm/overflow guards are needed and we can
// use v_exp_f32/v_log_f32 (base-2 hardware ops) directly.
__device__ __forceinline__ float softplus_f(float x) {
  const float LOG2E = 1.4426950408889634f;
  const float LN2 = 0.6931471805599453f;
  float t = __builtin_amdgcn_exp2f(-fabsf(x) * LOG2E);   // exp(-|x|)
  float y = __builtin_amdgcn_logf(1.0f + t);             // log2(1+t)
  return fmaxf(x, 0.0f) + LN2 * y;
}
__device__ __forceinline__ float sigmoid_f(float x) {
  return 1.0f / (1.0f + __builtin_amdgcn_exp2f(-x * 1.4426950408889634f));
}

// ---------------------------------------------------------------------------
// Prelude: e_id = MLP([3,3,1,-2]) and nu_red = sum(2*H*[1,2,1,-1]) are uniform
// over the batch; compute once into d_ws[0..1].
// ---------------------------------------------------------------------------
__global__ void icsa_prelude(const float* __restrict__ W0, const float* __restrict__ b0,
                             const float* __restrict__ W1, const float* __restrict__ b1,
                             const float* __restrict__ Wf, const float* __restrict__ bf,
                             float* __restrict__ cons) {
  if (threadIdx.x != 0 || blockIdx.x != 0) return;
  const float x[4] = {3.0f, 3.0f, 1.0f, -2.0f};
  float h0[8], s0[8];
  for (int mm = 0; mm < 8; ++mm) {
    float z = b0[mm];
    for (int k = 0; k < 4; ++k) z += x[k] * W0[k * 8 + mm];
    h0[mm] = softplus_f(z); s0[mm] = sigmoid_f(z);
  }
  float h1[8], s1[8];
  for (int mm = 0; mm < 8; ++mm) {
    float z = b1[mm];
    for (int k = 0; k < 8; ++k) z += h0[k] * W1[k * 8 + mm];
    h1[mm] = softplus_f(z); s1[mm] = sigmoid_f(z);
  }
  float zf = bf[0];
  for (int mm = 0; mm < 8; ++mm) zf += h1[mm] * Wf[mm];
  float e_id = softplus_f(zf);
  float sf = sigmoid_f(zf);
  float dz1[8];
  for (int mm = 0; mm < 8; ++mm) dz1[mm] = Wf[mm] * sf * s1[mm];
  float dz0[8];
  for (int k = 0; k < 8; ++k) {
    float a = 0.0f;
    for (int mm = 0; mm < 8; ++mm) a += W1[k * 8 + mm] * dz1[mm];
    dz0[k] = a * s0[k];
  }
  float H[4];
  for (int k = 0; k < 4; ++k) {
    float a = 0.0f;
    for (int mm = 0; mm < 8; ++mm) a += W0[k * 8 + mm] * dz0[mm];
    H[k] = a;
  }
  cons[0] = e_id;
  cons[1] = 2.0f * (H[0] + 2.0f * H[1] + H[2] - H[3]);
}

// ---------------------------------------------------------------------------
// Main kernel: 32 samples per wave per iteration.
//   - per-lane invariants from F
//   - MLP via V_WMMA_F32_16X16X4_F32 in transposed (h^T) form, 2 tiles of 16
//   - after each matrix layer the two tiles' D matrices are merged into a
//     full-wave packing (low lanes = tile0, high lanes = tile1) so every
//     softplus covers 32 useful values and lane l == sample base+l at the end.
//   - ALL __shfl calls are hoisted out of ternaries so the ds_bpermute ops
//     execute with full EXEC (bpermute reads from disabled lanes return 0).
// WMMA layouts (wave32):
//   A 16x4 f32:  lanes 0-15 = (K0,K1) of row=lane; lanes 16-31 = (K2,K3)
//   B 4x16  f32: V0 = {K0 row | K2 row}, V1 = {K1 row | K3 row}
//   C/D 16x16:   VGPR p: lanes 0-15 = row p, lanes 16-31 = row p+8
// ---------------------------------------------------------------------------
__global__ __launch_bounds__(256) void icsa_main(
    const float* __restrict__ Fp, const float* __restrict__ W0,
    const float* __restrict__ b0, const float* __restrict__ W1,
    const float* __restrict__ b1, const float* __restrict__ Wf,
    const float* __restrict__ bf, const float* __restrict__ cons,
    float* __restrict__ out, int B) {
  const int lane = threadIdx.x & 31;
  const int m = lane & 15;
  const int mh = m + 16;             // in [16,31] for every lane (no wrap)
  const bool lo = lane < 16;
  const bool valid = m < 8;          // MLP width 8; A rows 8..15 are zero pad
  const int kb = lo ? 0 : 2;

  // A matrices: transposed weights, zero-padded to 16 rows.
  v2f A0, A1a, A1b;
  A0.x  = valid ? W0[(kb + 0) * 8 + m] : 0.0f;
  A0.y  = valid ? W0[(kb + 1) * 8 + m] : 0.0f;
  A1a.x = valid ? W1[(kb + 0) * 8 + m] : 0.0f;
  A1a.y = valid ? W1[(kb + 1) * 8 + m] : 0.0f;
  A1b.x = valid ? W1[(kb + 4) * 8 + m] : 0.0f;
  A1b.y = valid ? W1[(kb + 5) * 8 + m] : 0.0f;

  // Biases as C matrices (rows 8..15 stay zero).
  v8f C0, C1;
#pragma unroll
  for (int p = 0; p < 8; ++p) {
    C0[p] = lo ? b0[p] : 0.0f;
    C1[p] = lo ? b1[p] : 0.0f;
  }

  float wf[8];
#pragma unroll
  for (int p = 0; p < 8; ++p) wf[p] = Wf[p];
  const float bff = bf[0];
  const float e_id = cons[0];
  const float nu_red = cons[1];

  const int waveId = (blockIdx.x * blockDim.x + threadIdx.x) >> 5;
  const int waveStride = (gridDim.x * blockDim.x) >> 5;

  for (int base = waveId * 32; base < B; base += waveStride * 32) {
    int i = base + lane;
    if (i >= B) i = B - 1;  // clamp: keeps EXEC all-ones for WMMA
    const float* f = Fp + (size_t)i * 9;
    const float f0 = f[0], f1 = f[1], f2 = f[2];
    const float f3 = f[3], f4 = f[4], f5 = f[5];
    const float f6 = f[6], f7 = f[7], f8 = f[8];

    // C = F^T F (symmetric, 6 unique entries)
    const float c00 = f0 * f0 + f3 * f3 + f6 * f6;
    const float c11 = f1 * f1 + f4 * f4 + f7 * f7;
    const float c22 = f2 * f2 + f5 * f5 + f8 * f8;
    const float c01 = f0 * f1 + f3 * f4 + f6 * f7;
    const float c02 = f0 * f2 + f3 * f5 + f6 * f8;
    const float c12 = f1 * f2 + f4 * f5 + f7 * f8;
    const float trC = c00 + c11 + c22;
    const float trC2 = c00 * c00 + c11 * c11 + c22 * c22 +
                       2.0f * (c01 * c01 + c02 * c02 + c12 * c12);
    const float I2 = 0.5f * (trC * trC - trC2);
    const float J = f0 * (f4 * f8 - f5 * f7) - f1 * (f3 * f8 - f5 * f6) +
                    f2 * (f3 * f7 - f4 * f6);
    const float inv0 = trC, inv1 = I2, inv2 = J * J, inv3 = -2.0f * J;

    // ---- layer 0 B matrices: inv^T (4x16) per tile ------------------------
    // Unconditional shuffles first (full EXEC), then lane-half selects.
    const float g2 = __shfl(inv2, m);    // for tile0 high half
    const float g3 = __shfl(inv3, m);
    const float g0 = __shfl(inv0, mh);   // for tile1 low half
    const float g1 = __shfl(inv1, mh);
    v2f Bm0, Bm1;
    Bm0.x = lo ? inv0 : g2;
    Bm0.y = lo ? inv1 : g3;
    Bm1.x = lo ? g0 : inv2;
    Bm1.y = lo ? g1 : inv3;

    v8f z0a = __builtin_amdgcn_wmma_f32_16x16x4_f32(
        false, A0, false, Bm0, (short)0, C0, false, false);
    v8f z0b = __builtin_amdgcn_wmma_f32_16x16x4_f32(
        false, A0, false, Bm1, (short)0, C0, false, false);

    // merge: S0[p] = softplus(row p), low lanes = tile0, high lanes = tile1
    float S0[8];
#pragma unroll
    for (int p = 0; p < 8; ++p) {
      const float zb = __shfl(z0b[p], m);        // unconditional
      const float mg = lo ? z0a[p] : zb;
      S0[p] = softplus_f(mg);
    }

    // ---- layer 1 B matrices from the packed S0 ----------------------------
    const float u2 = __shfl(S0[2], m);
    const float u3 = __shfl(S0[3], m);
    const float u6 = __shfl(S0[6], m);
    const float u7 = __shfl(S0[7], m);
    const float v0 = __shfl(S0[0], mh);
    const float v1 = __shfl(S0[1], mh);
    const float v4 = __shfl(S0[4], mh);
    const float v5 = __shfl(S0[5], mh);
    v2f B1a0, B1b0, B1a1, B1b1;
    B1a0.x = lo ? S0[0] : u2;  B1a0.y = lo ? S0[1] : u3;  // tile0 rows 0..3
    B1b0.x = lo ? S0[4] : u6;  B1b0.y = lo ? S0[5] : u7;  // tile0 rows 4..7
    B1a1.x = lo ? v0 : S0[2];  B1a1.y = lo ? v1 : S0[3];  // tile1 rows 0..3
    B1b1.x = lo ? v4 : S0[6];  B1b1.y = lo ? v5 : S0[7];  // tile1 rows 4..7

    v8f z1a = __builtin_amdgcn_wmma_f32_16x16x4_f32(
        false, A1a, false, B1a0, (short)0, C1, false, false);
    z1a = __builtin_amdgcn_wmma_f32_16x16x4_f32(
        false, A1b, false, B1b0, (short)0, z1a, false, false);
    v8f z1b = __builtin_amdgcn_wmma_f32_16x16x4_f32(
        false, A1a, false, B1a1, (short)0, C1, false, false);
    z1b = __builtin_amdgcn_wmma_f32_16x16x4_f32(
        false, A1b, false, B1b1, (short)0, z1b, false, false);

    // merge layer-1 output, then final dot: every lane is a distinct sample
    float zf = bff;
#pragma unroll
    for (int p = 0; p < 8; ++p) {
      const float zb = __shfl(z1b[p], m);        // unconditional
      const float mg = lo ? z1a[p] : zb;
      zf = fmaf(softplus_f(mg), wf[p], zf);
    }
    const float e = softplus_f(zf);

    const float d = J + 1.0f / J - 2.0f;
    const float r = e - e_id - nu_red * (J - 1.0f) + d * d;

    const int oi = base + lane;  // lane l == sample base+l after merging
    if (oi < B) out[oi] = r;
  }
}

extern "C" void kernel_launch(void* const* d_in, const int* in_sizes, int n_in,
                              void* d_out, int out_size, void* d_ws, size_t ws_size,
                              hipStream_t stream) {
  (void)n_in; (void)out_size; (void)ws_size;
  const float* Fp = (const float*)d_in[0];
  const float* W0 = (const float*)d_in[1];
  const float* b0 = (const float*)d_in[2];
  const float* W1 = (const float*)d_in[3];
  const float* b1 = (const float*)d_in[4];
  const float* Wf = (const float*)d_in[5];
  const float* bf = (const float*)d_in[6];
  float* out = (float*)d_out;
  float* cons = (float*)d_ws;  // [0]=e_id, [1]=nu_red
  const int B = in_sizes[0] / 9;

  icsa_prelude<<<1, 32, 0, stream>>>(W0, b0, W1, b1, Wf, bf, cons);

  const int threads = 256;  // 8 waves/block
  const int blocks = 2048;  // 16384 waves -> 8 grid-stride iters at B=2^22
  icsa_main<<<blocks, threads, 0, stream>>>(Fp, W0, b0, W1, b1, Wf, bf, cons,
                                            out, B);
}